// VGAEModel_59631325938137
// MI455X (gfx1250) — compile-verified
//
#include <hip/hip_runtime.h>
#include <hip/hip_bf16.h>
#include <math.h>

typedef __attribute__((ext_vector_type(2))) float v2f;
typedef __attribute__((ext_vector_type(8))) float v8f;

#define N_NODES 8192
#define N_EDGES 262144
#define IN_DIM  256
#define H1      64
#define H2      16

// ---------------- workspace layout (floats) ----------------
// [0]        norm_src / deg_src   : 8192
// [8192]     norm_dst / deg_dst   : 8192
// [16384]    hW0                  : 8192*64
// [540672]   agg1 (then h in-place): 8192*64
// [1064960]  hW1                  : 8192*16
// [1196032]  hW2                  : 8192*16
// [1327104]  agg2                 : 8192*16
// [1458176]  agg3                 : 8192*16
// [1589248]  z                    : 8192*16
// total 1720320 floats (~6.9 MB)
#define WS_TOTAL 1720320

__global__ void k_zero(float* __restrict__ p, int n) {
    int i = blockIdx.x * blockDim.x + threadIdx.x;
    if (i < n) p[i] = 0.0f;
}

__global__ void k_degree(const int* __restrict__ src, const int* __restrict__ dst,
                         float* __restrict__ deg_src, float* __restrict__ deg_dst) {
    int i = blockIdx.x * blockDim.x + threadIdx.x;
    if (i < N_EDGES) {
        atomicAdd(&deg_src[src[i]], 1.0f);
        atomicAdd(&deg_dst[dst[i]], 1.0f);
    }
}

__global__ void k_norm(float* __restrict__ deg_src, float* __restrict__ deg_dst) {
    int i = blockIdx.x * blockDim.x + threadIdx.x;
    if (i < N_NODES) {
        float a = deg_src[i]; a = a < 1.0f ? 1.0f : a;
        float b = deg_dst[i]; b = b < 1.0f ? 1.0f : b;
        deg_src[i] = rsqrtf(a);
        deg_dst[i] = rsqrtf(b);
    }
}

// GC1 dense part: hW0[row][j] = norm_src[row] * sum_k x[row][k] * W0[k][j]
// one block (64 threads) per row; row cached in LDS.
__global__ __launch_bounds__(H1)
void k_gc1(const float* __restrict__ x, const float* __restrict__ W0,
           const float* __restrict__ norm_src, float* __restrict__ hW0) {
    __shared__ float xs[IN_DIM];
    const int row = blockIdx.x;
    const int j   = threadIdx.x;
    const float* xr = x + (long long)row * IN_DIM;
    for (int k = j; k < IN_DIM; k += H1) xs[k] = xr[k];
    __syncthreads();
    float acc = 0.0f;
#pragma unroll 8
    for (int k = 0; k < IN_DIM; ++k) acc += xs[k] * W0[k * H1 + j];
    hW0[(long long)row * H1 + j] = acc * norm_src[row];
}

// edge scatter: agg[dst[e]][f] += h[src[e]][f]
template <int W>
__global__ void k_scatter1(const int* __restrict__ src, const int* __restrict__ dst,
                           const float* __restrict__ h, float* __restrict__ agg) {
    long long i = (long long)blockIdx.x * blockDim.x + threadIdx.x;
    if (i < (long long)N_EDGES * W) {
        int e = (int)(i / W);
        int f = (int)(i % W);
        atomicAdd(&agg[(long long)dst[e] * W + f], h[(long long)src[e] * W + f]);
    }
}

// h = relu(agg1 * norm_dst + b0), written in place over agg1
__global__ void k_gc1_finish(float* __restrict__ agg1, const float* __restrict__ norm_dst,
                             const float* __restrict__ b0) {
    int i = blockIdx.x * blockDim.x + threadIdx.x;
    if (i < N_NODES * H1) {
        int row = i / H1, j = i % H1;
        float v = agg1[i] * norm_dst[row] + b0[j];
        agg1[i] = v > 0.0f ? v : 0.0f;
    }
}

// fused GC2/GC3 dense parts: hW1 = (h*ns)@W1, hW2 = (h*ns)@W2
__global__ void k_gc23(const float* __restrict__ h, const float* __restrict__ W1,
                       const float* __restrict__ W2, const float* __restrict__ norm_src,
                       float* __restrict__ hW1, float* __restrict__ hW2) {
    int i = blockIdx.x * blockDim.x + threadIdx.x;
    if (i < N_NODES * H2) {
        int row = i >> 4, j = i & 15;
        const float* hr = h + (long long)row * H1;
        float ns = norm_src[row];
        float a1 = 0.0f, a2 = 0.0f;
#pragma unroll 8
        for (int k = 0; k < H1; ++k) {
            float hv = hr[k];
            a1 += hv * W1[k * H2 + j];
            a2 += hv * W2[k * H2 + j];
        }
        hW1[i] = a1 * ns;
        hW2[i] = a2 * ns;
    }
}

// fused edge scatter for both mean/log_std aggregations
__global__ void k_scatter23(const int* __restrict__ src, const int* __restrict__ dst,
                            const float* __restrict__ h1, const float* __restrict__ h2,
                            float* __restrict__ agg2, float* __restrict__ agg3) {
    long long i = (long long)blockIdx.x * blockDim.x + threadIdx.x;
    if (i < (long long)N_EDGES * H2) {
        int e = (int)(i >> 4);
        int f = (int)(i & 15);
        long long so = (long long)src[e] * H2 + f;
        long long dofs = (long long)dst[e] * H2 + f;
        atomicAdd(&agg2[dofs], h1[so]);
        atomicAdd(&agg3[dofs], h2[so]);
    }
}

// z = (agg2*nd + b1) + noise * exp(agg3*nd + b2)
__global__ void k_z(const float* __restrict__ agg2, const float* __restrict__ agg3,
                    const float* __restrict__ norm_dst, const float* __restrict__ b1,
                    const float* __restrict__ b2, const float* __restrict__ noise,
                    float* __restrict__ z) {
    int i = blockIdx.x * blockDim.x + threadIdx.x;
    if (i < N_NODES * H2) {
        int row = i >> 4, j = i & 15;
        float nd = norm_dst[row];
        float mean = agg2[i] * nd + b1[j];
        float lstd = agg3[i] * nd + b2[j];
        z[i] = mean + noise[i] * __expf(lstd);
    }
}

// Decoder: out = sigmoid(z @ z^T), 16x16 tile per wave via V_WMMA_F32_16X16X4_F32.
// 8 waves / block, 512x512 tiles total. z (512KB) lives in L2; output is a
// 256MB write-once stream -> non-temporal stores. Sigmoid epilogue uses a
// single v_rcp_f32 (not IEEE divide) so VALU stays under the HBM store roofline.
__global__ __launch_bounds__(256)
void k_decoder(const float* __restrict__ z, float* __restrict__ out) {
    const int wave = threadIdx.x >> 5;
    const int lane = threadIdx.x & 31;
    const int tile = blockIdx.x * 8 + wave;
    const int tr = tile >> 9;          // tile row   (0..511)
    const int tc = tile & 511;         // tile col   (0..511)
    const int row0 = tr << 4;
    const int col0 = tc << 4;
    const int tl = lane & 15;
    const int hi = (lane >> 4) & 1;    // lanes 16-31 carry K+2,K+3 (A/B) and M+8 (C/D)

    // per-lane base pointers into z rows (A: output row, B: output col via z^T)
    const float* arow = z + ((long long)(row0 + tl) * H2 + 2 * hi);
    const float* brow = z + ((long long)(col0 + tl) * H2 + 2 * hi);

    v8f c = {};
#pragma unroll
    for (int i = 0; i < 4; ++i) {
        v2f a, b;
        a.x = arow[4 * i];     a.y = arow[4 * i + 1];
        b.x = brow[4 * i];     b.y = brow[4 * i + 1];
        // D = A(16x4) * B(4x16) + C   (fp32, exact match to reference math)
        c = __builtin_amdgcn_wmma_f32_16x16x4_f32(
                /*neg_a=*/false, a, /*neg_b=*/false, b,
                /*c_mod=*/(short)0, c, /*reuse_a=*/false, /*reuse_b=*/false);
    }

    // C/D layout: VGPR j -> M = j + 8*hi, N = tl.  Fast sigmoid + streaming store.
    float* o = out + (long long)(row0 + hi * 8) * N_NODES + (col0 + tl);
#pragma unroll
    for (int j = 0; j < 8; ++j) {
        float e = __expf(-c[j]);                       // v_exp_f32 (+ ln2 scale)
        float v = __builtin_amdgcn_rcpf(1.0f + e);     // single v_rcp_f32
        __builtin_nontemporal_store(v, o + (long long)j * N_NODES);
    }
}

extern "C" void kernel_launch(void* const* d_in, const int* in_sizes, int n_in,
                              void* d_out, int out_size, void* d_ws, size_t ws_size,
                              hipStream_t stream) {
    const float* features = (const float*)d_in[0];
    const int*   src      = (const int*)  d_in[1];
    const int*   dst      = (const int*)  d_in[2];
    const float* noise    = (const float*)d_in[3];
    const float* W0       = (const float*)d_in[4];
    const float* b0       = (const float*)d_in[5];
    const float* W1       = (const float*)d_in[6];
    const float* b1       = (const float*)d_in[7];
    const float* W2       = (const float*)d_in[8];
    const float* b2       = (const float*)d_in[9];
    float* out = (float*)d_out;

    float* ws       = (float*)d_ws;
    float* norm_src = ws;                // 8192
    float* norm_dst = ws + 8192;         // 8192
    float* hW0      = ws + 16384;        // 8192*64
    float* agg1     = ws + 540672;       // 8192*64  (becomes h after relu)
    float* hW1      = ws + 1064960;      // 8192*16
    float* hW2      = ws + 1196032;      // 8192*16
    float* agg2     = ws + 1327104;      // 8192*16
    float* agg3     = ws + 1458176;      // 8192*16
    float* zbuf     = ws + 1589248;      // 8192*16

    // 1. zero accumulator regions (whole used ws for simplicity)
    k_zero<<<(WS_TOTAL + 255) / 256, 256, 0, stream>>>(ws, WS_TOTAL);

    // 2. degrees + norms
    k_degree<<<(N_EDGES + 255) / 256, 256, 0, stream>>>(src, dst, norm_src, norm_dst);
    k_norm<<<(N_NODES + 255) / 256, 256, 0, stream>>>(norm_src, norm_dst);

    // 3. GC layer 1
    k_gc1<<<N_NODES, H1, 0, stream>>>(features, W0, norm_src, hW0);
    k_scatter1<H1><<<(N_EDGES * H1) / 256, 256, 0, stream>>>(src, dst, hW0, agg1);
    k_gc1_finish<<<(N_NODES * H1) / 256, 256, 0, stream>>>(agg1, norm_dst, b0);

    // 4. GC layers 2+3 (fused)
    k_gc23<<<(N_NODES * H2) / 256, 256, 0, stream>>>(agg1, W1, W2, norm_src, hW1, hW2);
    k_scatter23<<<(N_EDGES * H2) / 256, 256, 0, stream>>>(src, dst, hW1, hW2, agg2, agg3);

    // 5. reparameterize
    k_z<<<(N_NODES * H2) / 256, 256, 0, stream>>>(agg2, agg3, norm_dst, b1, b2, noise, zbuf);

    // 6. WMMA decoder: 512*512 tiles, 8 tiles (waves) per block
    k_decoder<<<(512 * 512) / 8, 256, 0, stream>>>(zbuf, out);
}